// LLFullObjectCondensation_49598282334785
// MI455X (gfx1250) — compile-verified
//
#include <hip/hip_runtime.h>
#include <hip/hip_bf16.h>
#include <cstdint>
#include <math.h>

#define NPTS 60000
#define KCL  800
#define QMIN 0.1f

typedef __attribute__((ext_vector_type(2))) float v2f;
typedef __attribute__((ext_vector_type(8))) float v8f;

__device__ __forceinline__ float wave_sum32(float v) {
#pragma unroll
  for (int off = 16; off > 0; off >>= 1) v += __shfl_xor(v, off, 32);
  return v;
}

// fast sqrt: raw v_sqrt_f32 (~1 ulp). Inputs here are >= 1e-6, far from
// denormals, so the precise-sqrt expansion is wasted VALU work.
__device__ __forceinline__ float fsqrt_fast(float x) {
  return __builtin_amdgcn_sqrtf(x);
}

// scal[] slots: 0=rep(all-K) 1=att 2=beta 3=n_valid 4=noise_b 5=n_noise
//               6=payload_total 7=rep_self_correction
#define S_REP   0
#define S_ATT   1
#define S_BETA  2
#define S_NV    3
#define S_NOI   4
#define S_NN    5
#define S_PAY   6
#define S_REPC  7

// ---------------- K1: zero accumulators in workspace ----------------
__global__ void k_init(float* fzero, int nfloats, unsigned long long* packed) {
  int i = blockIdx.x * blockDim.x + threadIdx.x;
  if (i < nfloats) fzero[i] = 0.0f;
  if (i < KCL) packed[i] = 0ull;
}

// ---------------- K2: per-point pass A ----------------
// b, q; segment counts, dep energy, packed argmax(b, idx); noise sums.
__global__ void k_point_a(const float* __restrict__ beta,
                          const int* __restrict__ t_idx,
                          const float* __restrict__ hit_energy,
                          float* __restrict__ q, float* __restrict__ bArr,
                          float* __restrict__ counts, float* __restrict__ dep,
                          unsigned long long* __restrict__ packed,
                          float* __restrict__ scal) {
  int i = blockIdx.x * blockDim.x + threadIdx.x;
  float noise_b = 0.f, noise_c = 0.f;
  if (i < NPTS) {
    float b = beta[i];
    b = fminf(fmaxf(b, 1e-6f), 1.f - 1e-6f);
    float at = atanhf(b);
    q[i] = at * at + QMIN;
    bArr[i] = b;
    int tr = t_idx[i];
    if (tr >= 0) {
      atomicAdd(&counts[tr], 1.0f);
      atomicAdd(&dep[tr], hit_energy[i]);
      unsigned long long key =
          (((unsigned long long)__float_as_uint(b)) << 32) | (unsigned)i;
      atomicMax(&packed[tr], key);
    } else {
      noise_b = b; noise_c = 1.0f;
    }
  }
  noise_b = wave_sum32(noise_b);
  noise_c = wave_sum32(noise_c);
  if ((threadIdx.x & 31) == 0) {
    if (noise_b != 0.f) atomicAdd(&scal[S_NOI], noise_b);
    if (noise_c != 0.f) atomicAdd(&scal[S_NN], noise_c);
  }
}

// ---------------- K3: per-cluster alpha extraction ----------------
// bvec[k] = (-2x, -2y, 1, |x|^2) so that a(i)·bvec(k) = d2(i,k)
__global__ void k_cluster(const float* __restrict__ counts,
                          const unsigned long long* __restrict__ packed,
                          const float* __restrict__ ccoords,
                          const float* __restrict__ q,
                          const float* __restrict__ bArr,
                          float* __restrict__ qa, float* __restrict__ validf,
                          float4* __restrict__ bvec, float* __restrict__ scal) {
  int k = blockIdx.x * blockDim.x + threadIdx.x;
  float betaTerm = 0.f, vTerm = 0.f;
  if (k < KCL) {
    bool valid = counts[k] > 0.f;
    unsigned long long pk = packed[k];
    int ai = valid ? (int)(pk & 0xffffffffull) : 0;
    float x0 = ccoords[2 * ai], x1 = ccoords[2 * ai + 1];
    qa[k] = valid ? q[ai] : 0.f;
    validf[k] = valid ? 1.f : 0.f;
    bvec[k] = make_float4(-2.f * x0, -2.f * x1, 1.f, x0 * x0 + x1 * x1);
    if (valid) { betaTerm = 1.f - bArr[ai]; vTerm = 1.f; }
  }
  betaTerm = wave_sum32(betaTerm);
  vTerm = wave_sum32(vTerm);
  if ((threadIdx.x & 31) == 0) {
    if (betaTerm != 0.f) atomicAdd(&scal[S_BETA], betaTerm);
    if (vTerm != 0.f) atomicAdd(&scal[S_NV], vTerm);
  }
}

// ---------------- K4: N x K repulsion via V_WMMA_F32_16X16X4_F32 ----------------
// Each wave32 owns 16 points (rows), loops over 50 k-tiles of 16 alphas.
// Alpha table (bvec/qa, 16 KB) staged in LDS once per workgroup. The
// (1-M) self-cluster mask is NOT applied here: the own-cluster term is
// subtracted in the O(N) pass k_point_b, keeping this loop mask-free.
__global__ void __launch_bounds__(256) k_rep(const float* __restrict__ ccoords,
                                             const float* __restrict__ q,
                                             const float* __restrict__ qa,
                                             const float4* __restrict__ bvec,
                                             float* __restrict__ scal) {
  __shared__ float4 s_bvec[KCL];
  __shared__ float  s_qa[KCL];
  for (int t = threadIdx.x; t < KCL; t += blockDim.x) {
    s_bvec[t] = bvec[t];
    s_qa[t] = qa[t];
  }
  __syncthreads();

  const int lane = threadIdx.x & 31;
  const int wave = (blockIdx.x * blockDim.x + threadIdx.x) >> 5;
  const int numTiles = NPTS / 16;  // 3750
  if (wave >= numTiles) return;    // wave-uniform: EXEC stays all-ones for WMMA
  const int i0 = wave * 16;
  const int m = lane & 15;
  const int h = lane >> 4;

  // A operand: row m = point i0+m; lane (m,h) holds A[m][2h], A[m][2h+1]
  const int pi = i0 + m;
  const float cx = ccoords[2 * pi], cy = ccoords[2 * pi + 1];
  const float cn2 = cx * cx + cy * cy;
  v2f A;
  A[0] = h ? cn2 : cx;
  A[1] = h ? 1.0f : cy;

  // rows this lane post-processes from D: M = 8h + v
  float qrow[8];
  const int rbase = i0 + 8 * h;
#pragma unroll
  for (int v = 0; v < 8; ++v) qrow[v] = q[rbase + v];

  float acc = 0.f;
  for (int kt = 0; kt < KCL / 16; ++kt) {
    const int kn = kt * 16 + m;     // this lane's alpha column N = m
    const float4 bv = s_bvec[kn];
    v2f B;                           // lane (m,h) holds B[2h][m], B[2h+1][m]
    B[0] = h ? bv.z : bv.x;
    B[1] = h ? bv.w : bv.y;
    const float qak = s_qa[kn];
    v8f C = {};
    v8f D = __builtin_amdgcn_wmma_f32_16x16x4_f32(
        false, A, false, B, (short)0, C, false, false);
    float s = 0.f;
#pragma unroll
    for (int v = 0; v < 8; ++v) {
      float t = 1.0f - fsqrt_fast(D[v] + 1e-6f);
      t = fmaxf(t, 0.0f);
      s = fmaf(qrow[v], t, s);       // factor q_alpha out of the row sum
    }
    acc = fmaf(qak, s, acc);
  }
  acc = wave_sum32(acc);
  if (lane == 0) atomicAdd(&scal[S_REP], acc);
}

// ---------------- K5: per-point pass B: attraction + payload + rep-corr ----------------
__global__ void k_point_b(const float* __restrict__ ccoords,
                          const float* __restrict__ q,
                          const float* __restrict__ bArr,
                          const int* __restrict__ t_idx,
                          const float* __restrict__ pred_energy,
                          const float* __restrict__ pred_pos,
                          const float* __restrict__ pred_time,
                          const float* __restrict__ t_energy,
                          const float* __restrict__ t_pos,
                          const float* __restrict__ t_time,
                          const float* __restrict__ dep,
                          const float* __restrict__ qa,
                          const float4* __restrict__ bvec,
                          float* __restrict__ bsum, float* __restrict__ psum,
                          float* __restrict__ scal) {
  int i = blockIdx.x * blockDim.x + threadIdx.x;
  float att = 0.f, repc = 0.f;
  if (i < NPTS) {
    int tr = t_idx[i];
    if (tr >= 0) {
      float cx = ccoords[2 * i], cy = ccoords[2 * i + 1];
      float4 bv = bvec[tr];
      float d2 = bv.x * cx + bv.y * cy + bv.z * (cx * cx + cy * cy) + bv.w;
      float qq = q[i] * qa[tr];
      att = qq * d2;
      // own-cluster repulsion term (excluded by (1-M) in the reference),
      // subtracted from the unmasked N x K sum computed by k_rep.
      float ts = 1.0f - fsqrt_fast(d2 + 1e-6f);
      repc = qq * fmaxf(ts, 0.0f);
      // payload (noise rows have bw = 0 in the reference, so skipped entirely)
      float dh = dep[tr];
      float corr = t_energy[i] / fmaxf(dh, 1e-6f);
      float de = pred_energy[i] - corr;
      float eloss = de * de / (corr + 1.0f);
      float dx = pred_pos[2 * i] - t_pos[2 * i];
      float dy = pred_pos[2 * i + 1] - t_pos[2 * i + 1];
      float r = fsqrt_fast(dx * dx + dy * dy + 0.01f);
      float ploss = (r < 10.f) ? r * r : (100.f + 20.f * (r - 10.f));  // huber d=10
      float dt = pred_time[i] - t_time[i];
      float tloss = dt * dt;
      float te = t_energy[i];
      float ew = (te > 10.f) ? 1.f : (te - 0.5f) / 9.5f;
      ew = fmaxf(ew, 0.f);
      float payload = (eloss + ploss + tloss) * ew;
      float bw = bArr[i];
      atomicAdd(&bsum[tr], bw);
      atomicAdd(&psum[tr], bw * payload);
    }
  }
  att = wave_sum32(att);
  repc = wave_sum32(repc);
  if ((threadIdx.x & 31) == 0) {
    if (att != 0.f) atomicAdd(&scal[S_ATT], att);
    if (repc != 0.f) atomicAdd(&scal[S_REPC], repc);
  }
}

// ---------------- K6: per-cluster payload normalization ----------------
__global__ void k_payload(const float* __restrict__ bsum,
                          const float* __restrict__ psum,
                          const float* __restrict__ validf,
                          float* __restrict__ scal) {
  int k = blockIdx.x * blockDim.x + threadIdx.x;
  float po = 0.f;
  if (k < KCL && validf[k] > 0.f) po = psum[k] / fmaxf(bsum[k], 1e-6f);
  po = wave_sum32(po);
  if ((threadIdx.x & 31) == 0 && po != 0.f) atomicAdd(&scal[S_PAY], po);
}

// ---------------- K7: final combine ----------------
__global__ void k_final(const float* __restrict__ scal, float* __restrict__ out) {
  if (threadIdx.x == 0 && blockIdx.x == 0) {
    float nv = fmaxf(scal[S_NV], 1.0f);
    float nn = fmaxf(scal[S_NN], 1.0f);
    const float Nf = (float)NPTS;
    float rep = scal[S_REP] - scal[S_REPC];  // unmasked sum minus own-cluster
    // L_att + L_rep + L_beta + L_noise + L_payload (all weights == 1)
    out[0] = scal[S_ATT] / Nf + rep / Nf + scal[S_BETA] / nv +
             scal[S_NOI] / nn + scal[S_PAY] / nv;
  }
}

extern "C" void kernel_launch(void* const* d_in, const int* in_sizes, int n_in,
                              void* d_out, int out_size, void* d_ws,
                              size_t ws_size, hipStream_t stream) {
  (void)in_sizes; (void)n_in; (void)out_size; (void)ws_size;
  // Inputs in setup_inputs() order:
  const float* beta        = (const float*)d_in[0];
  const float* ccoords     = (const float*)d_in[1];
  const float* pred_energy = (const float*)d_in[2];
  const float* pred_pos    = (const float*)d_in[3];
  const float* pred_time   = (const float*)d_in[4];
  const float* hit_energy  = (const float*)d_in[5];
  const float* t_energy    = (const float*)d_in[6];
  const float* t_pos       = (const float*)d_in[7];
  const float* t_time      = (const float*)d_in[8];
  const int*   t_idx       = (const int*)d_in[9];
  float* out = (float*)d_out;

  // Workspace layout
  float* q      = (float*)d_ws;          // N
  float* bArr   = q + NPTS;              // N
  float* counts = bArr + NPTS;           // K      (zeroed)
  float* dep    = counts + KCL;          // K+1    (zeroed)
  float* qa     = dep + (KCL + 1);       // K      (zeroed, then written)
  float* validf = qa + KCL;              // K      (zeroed, then written)
  float* bsum   = validf + KCL;          // K      (zeroed)
  float* psum   = bsum + KCL;            // K      (zeroed)
  float* scal   = psum + KCL;            // 16 scalars (zeroed)
  uintptr_t p = (uintptr_t)(scal + 16);
  p = (p + 15) & ~(uintptr_t)15;
  float4* bvec = (float4*)p;             // K float4
  unsigned long long* packed = (unsigned long long*)(bvec + KCL);  // K (zeroed)

  const int nzero = KCL + (KCL + 1) + KCL + KCL + KCL + KCL + 16;  // from counts

  k_init<<<(nzero + 255) / 256, 256, 0, stream>>>(counts, nzero, packed);

  k_point_a<<<(NPTS + 255) / 256, 256, 0, stream>>>(
      beta, t_idx, hit_energy, q, bArr, counts, dep, packed, scal);

  k_cluster<<<(KCL + 255) / 256, 256, 0, stream>>>(
      counts, packed, ccoords, q, bArr, qa, validf, bvec, scal);

  const int numWaves = NPTS / 16;  // 3750 waves, 8 waves / 256-thread block
  k_rep<<<(numWaves + 7) / 8, 256, 0, stream>>>(ccoords, q, qa, bvec, scal);

  k_point_b<<<(NPTS + 255) / 256, 256, 0, stream>>>(
      ccoords, q, bArr, t_idx, pred_energy, pred_pos, pred_time, t_energy,
      t_pos, t_time, dep, qa, bvec, bsum, psum, scal);

  k_payload<<<(KCL + 255) / 256, 256, 0, stream>>>(bsum, psum, validf, scal);

  k_final<<<1, 1, 0, stream>>>(scal, out);
}